// Coords2RMSD_59210419143181
// MI455X (gfx1250) — compile-verified
//
#include <hip/hip_runtime.h>
#include <hip/hip_bf16.h>

typedef __attribute__((ext_vector_type(2))) float v2f;
typedef __attribute__((ext_vector_type(8))) float v8f;

#define BDIM 256
#define S_SLICES 8
#define NQ 16

// ---------------------------------------------------------------------------
// Kernel 1: streaming raw-moment reduction. grid = B*S_SLICES blocks.
// Quantities: [0..2]=Sum x, [3..5]=Sum y, [6..14]=Sum x_i*y_j (row major),
//             [15] = Sum |x|^2 + |y|^2.  All masked (a < num_atoms only).
// ---------------------------------------------------------------------------
__global__ void __launch_bounds__(BDIM)
k1_partials(const float* __restrict__ x, const float* __restrict__ y,
            const int* __restrict__ num_atoms, float* __restrict__ ws,
            int N /* MAX_ATOMS */) {
  const int bs = blockIdx.x;
  const int b  = bs / S_SLICES;
  const int s  = bs % S_SLICES;
  const int n  = num_atoms[b];
  const int chunk = (N + S_SLICES - 1) / S_SLICES;
  const int a0 = s * chunk;
  int a1 = a0 + chunk;
  if (a1 > n) a1 = n;

  const float* __restrict__ xb = x + (size_t)b * 3u * (size_t)N;
  const float* __restrict__ yb = y + (size_t)b * 3u * (size_t)N;

  float acc[NQ];
#pragma unroll
  for (int q = 0; q < NQ; ++q) acc[q] = 0.0f;

  for (int a = a0 + (int)threadIdx.x; a < a1; a += BDIM) {
    const float* xp = xb + 3 * a;
    const float* yp = yb + 3 * a;
    float x0 = xp[0], x1 = xp[1], x2 = xp[2];
    float y0 = yp[0], y1 = yp[1], y2 = yp[2];
    acc[0] += x0;  acc[1] += x1;  acc[2] += x2;
    acc[3] += y0;  acc[4] += y1;  acc[5] += y2;
    acc[6]  += x0 * y0; acc[7]  += x0 * y1; acc[8]  += x0 * y2;
    acc[9]  += x1 * y0; acc[10] += x1 * y1; acc[11] += x1 * y2;
    acc[12] += x2 * y0; acc[13] += x2 * y1; acc[14] += x2 * y2;
    acc[15] += x0 * x0 + x1 * x1 + x2 * x2 + y0 * y0 + y1 * y1 + y2 * y2;
  }

  // wave32 intra-wave tree reduction
#pragma unroll
  for (int off = 16; off > 0; off >>= 1) {
#pragma unroll
    for (int q = 0; q < NQ; ++q)
      acc[q] += __shfl_down(acc[q], off, 32);
  }

  __shared__ float red[BDIM / 32][NQ];
  const int wave = threadIdx.x >> 5;
  const int lane = threadIdx.x & 31;
  if (lane == 0) {
#pragma unroll
    for (int q = 0; q < NQ; ++q) red[wave][q] = acc[q];
  }
  __syncthreads();

  if (threadIdx.x < NQ) {
    float t = 0.0f;
#pragma unroll
    for (int w = 0; w < BDIM / 32; ++w) t += red[w][threadIdx.x];
    ws[(size_t)bs * NQ + threadIdx.x] = t;
  }
}

// ---------------------------------------------------------------------------
// Kernel 2: one wave (32 threads) per protein.
// Reduce the 8 slice-partials of the 16 quantities with two chained
// V_WMMA_F32_16X16X4_F32 ops (B = ones => D[q][*] = sum_k A[q][k]), then
// lane 0 runs the f64 closed-form 3x3 Kabsch finalize.
//
// A-matrix 16x4 f32 ISA layout: lanes 0-15 hold M=0..15 with VGPR0=K0,
// VGPR1=K1; lanes 16-31 hold the same M with VGPR0=K2, VGPR1=K3.
// ---------------------------------------------------------------------------
__global__ void __launch_bounds__(32)
k2_finalize(const float* __restrict__ ws, const int* __restrict__ num_atoms,
            float* __restrict__ out) {
  const int b    = blockIdx.x;
  const int lane = threadIdx.x;      // 0..31, EXEC all ones before the WMMAs
  const int m    = lane & 15;
  const bool hi  = lane >= 16;
  const float* base = ws + (size_t)b * S_SLICES * NQ;

  // first WMMA: slices 0..3 ; second WMMA: slices 4..7
  const int kA = hi ? 2 : 0;
  const int kB = hi ? 6 : 4;
  v2f a01, a23;
  a01.x = base[(kA + 0) * NQ + m];
  a01.y = base[(kA + 1) * NQ + m];
  a23.x = base[(kB + 0) * NQ + m];
  a23.y = base[(kB + 1) * NQ + m];
  v2f ones;
  ones.x = 1.0f; ones.y = 1.0f;     // all-ones B matrix: layout invariant

  v8f c = {};
  c = __builtin_amdgcn_wmma_f32_16x16x4_f32(false, a01, false, ones,
                                            (short)0, c, false, false);
  c = __builtin_amdgcn_wmma_f32_16x16x4_f32(false, a23, false, ones,
                                            (short)0, c, false, false);

  // D layout: VGPR j holds rows M=j (lanes 0-15) and M=j+8 (lanes 16-31).
  __shared__ float sums[NQ];
  if (lane == 0) {
#pragma unroll
    for (int i = 0; i < 8; ++i) sums[i] = c[i];       // quantities 0..7 (N=0)
  }
  if (lane == 16) {
#pragma unroll
    for (int i = 0; i < 8; ++i) sums[8 + i] = c[i];   // quantities 8..15
  }
  __syncthreads();

  if (lane == 0) {
    const double n = (double)num_atoms[b];
    double Sx[3] = { sums[0], sums[1], sums[2] };
    double Sy[3] = { sums[3], sums[4], sums[5] };
    double Cr[9];
#pragma unroll
    for (int i = 0; i < 9; ++i) Cr[i] = sums[6 + i];
    const double E = sums[15];

    double xm[3], ym[3];
#pragma unroll
    for (int i = 0; i < 3; ++i) { xm[i] = Sx[i] / n; ym[i] = Sy[i] / n; }

    // centered cross-covariance and E0
    double C[3][3];
#pragma unroll
    for (int i = 0; i < 3; ++i)
#pragma unroll
      for (int j = 0; j < 3; ++j)
        C[i][j] = Cr[3 * i + j] - n * xm[i] * ym[j];

    const double E0 = E - n * (xm[0] * xm[0] + xm[1] * xm[1] + xm[2] * xm[2] +
                               ym[0] * ym[0] + ym[1] * ym[1] + ym[2] * ym[2]);

    const double detC =
        C[0][0] * (C[1][1] * C[2][2] - C[1][2] * C[2][1]) -
        C[0][1] * (C[1][0] * C[2][2] - C[1][2] * C[2][0]) +
        C[0][2] * (C[1][0] * C[2][1] - C[1][1] * C[2][0]);

    // K = C^T C (symmetric PSD); singular values of C = sqrt(eig(K))
    const double K00 = C[0][0]*C[0][0] + C[1][0]*C[1][0] + C[2][0]*C[2][0];
    const double K11 = C[0][1]*C[0][1] + C[1][1]*C[1][1] + C[2][1]*C[2][1];
    const double K22 = C[0][2]*C[0][2] + C[1][2]*C[1][2] + C[2][2]*C[2][2];
    const double K01 = C[0][0]*C[0][1] + C[1][0]*C[1][1] + C[2][0]*C[2][1];
    const double K02 = C[0][0]*C[0][2] + C[1][0]*C[1][2] + C[2][0]*C[2][2];
    const double K12 = C[0][1]*C[0][2] + C[1][1]*C[1][2] + C[2][1]*C[2][2];

    // closed-form eigenvalues of a symmetric 3x3 (trigonometric method)
    const double q3 = (K00 + K11 + K22) / 3.0;
    const double p1 = K01 * K01 + K02 * K02 + K12 * K12;
    const double d0 = K00 - q3, d1 = K11 - q3, d2 = K22 - q3;
    const double p2 = d0 * d0 + d1 * d1 + d2 * d2 + 2.0 * p1;
    double e1, e2, e3;
    if (p2 <= 1e-300) {
      e1 = e2 = e3 = q3;
    } else {
      const double p  = sqrt(p2 / 6.0);
      const double ip = 1.0 / p;
      const double B00 = d0 * ip, B11 = d1 * ip, B22 = d2 * ip;
      const double B01 = K01 * ip, B02 = K02 * ip, B12 = K12 * ip;
      double detB = B00 * (B11 * B22 - B12 * B12) -
                    B01 * (B01 * B22 - B12 * B02) +
                    B02 * (B01 * B12 - B11 * B02);
      double r = 0.5 * detB;
      r = fmin(1.0, fmax(-1.0, r));
      const double phi = acos(r) / 3.0;
      e1 = q3 + 2.0 * p * cos(phi);
      e3 = q3 + 2.0 * p * cos(phi + 2.0943951023931953);  // + 2*pi/3
      e2 = 3.0 * q3 - e1 - e3;                             // e1 >= e2 >= e3
    }
    const double s1 = sqrt(fmax(e1, 0.0));
    const double s2 = sqrt(fmax(e2, 0.0));
    const double s3 = sqrt(fmax(e3, 0.0));
    const double dsgn  = (detC < 0.0) ? -1.0 : 1.0;  // reflection correction
    const double sum_s = s1 + s2 + dsgn * s3;        // s3 = smallest

    const double msd = fmax(E0 - 2.0 * sum_s, 0.0) / n;
    out[b] = (float)sqrt(msd + 1e-8);
  }
}

// ---------------------------------------------------------------------------
extern "C" void kernel_launch(void* const* d_in, const int* in_sizes, int n_in,
                              void* d_out, int out_size, void* d_ws, size_t ws_size,
                              hipStream_t stream) {
  const float* x  = (const float*)d_in[0];   // input  [B, 3*N] f32
  const float* y  = (const float*)d_in[1];   // target [B, 3*N] f32
  const int*   na = (const int*)d_in[2];     // num_atoms [B] i32
  float* out = (float*)d_out;                // [B] f32
  float* wsf = (float*)d_ws;                 // B*S_SLICES*NQ floats (64 KB)

  const int B = in_sizes[2];
  const int N = in_sizes[0] / (3 * B);       // MAX_ATOMS

  k1_partials<<<B * S_SLICES, BDIM, 0, stream>>>(x, y, na, wsf, N);
  k2_finalize<<<B, 32, 0, stream>>>(wsf, na, out);
}